// Net_TRANSFORMER_85667417686730
// MI455X (gfx1250) — compile-verified
//
#include <hip/hip_runtime.h>
#include <math.h>

typedef __attribute__((ext_vector_type(2))) float v2f;
typedef __attribute__((ext_vector_type(8))) float v8f;

#define NNODES 20000
#define NEDGES 320000
#define NHEADS 4
#define SM_EPS 1e-16f
#define BN_EPS 1e-5f

// ---------------- device helpers ----------------

__device__ __forceinline__ void atomicMaxF(float* addr, float val) {
  // standard sign-split trick; works with init = -inf
  if (val >= 0.0f) atomicMax((int*)addr, __float_as_int(val));
  else             atomicMin((unsigned int*)addr, __float_as_uint(val));
}

__device__ __forceinline__ float sigmoidf_(float x) {
  return 1.0f / (1.0f + __expf(-x));
}

// ---------------- kernels ----------------

__global__ void fill_kernel(float* __restrict__ p, float v, int n) {
  int i = blockIdx.x * blockDim.x + threadIdx.x;
  if (i < n) p[i] = v;
}

// Y[NNODES, kout] = X[NNODES, KIN] @ W[KIN, kout] + bias
// One wave computes one 16x16 output tile with V_WMMA_F32_16X16X4_F32 (exact fp32).
// A 16x4 layout: lanes 0-15: M=lane, (K=k0,k0+1); lanes 16-31: M=lane-16, (K=k0+2,k0+3).
// B 4x16 mirrors A with N across lanes. C/D: VGPR r -> row r (lanes 0-15) / r+8 (lanes 16-31).
// KIN is a template constant; all loads in the hot loop are UNCONDITIONAL (no EXEC-masked
// loads): ragged kout (8, 2 in layer 4) is handled by clamped addresses + v_cndmask selects.
template <int KIN>
__global__ void gemm_bias_wmma(const float* __restrict__ X,
                               const float* __restrict__ W,
                               const float* __restrict__ bias,
                               float* __restrict__ Y,
                               int kout, int tilesN, int numTiles) {
  int wave = (int)((blockIdx.x * blockDim.x + threadIdx.x) >> 5);
  int lane = threadIdx.x & 31;
  if (wave >= numTiles) return;          // wave-uniform: EXEC stays all-ones below
  int tileM = wave / tilesN;
  int tileN = wave - tileM * tilesN;
  int half = lane >> 4;                  // 0: lanes 0-15, 1: lanes 16-31
  int l15  = lane & 15;
  int row = tileM * 16 + l15;            // A row (M); always < NNODES (20000 % 16 == 0)
  int col = tileN * 16 + l15;            // B / output column (N)
  bool colok = col < kout;
  int colc = colok ? col : (kout - 1);   // clamped, always in-bounds

  const float* __restrict__ xrow = X + (size_t)row * KIN;
  v8f acc = {0.f, 0.f, 0.f, 0.f, 0.f, 0.f, 0.f, 0.f};

  constexpr int KFULL = KIN & ~3;        // full 4-wide K steps: no range checks at all
  for (int k0 = 0; k0 < KFULL; k0 += 4) {
    int ka = k0 + 2 * half;
    v2f a, b;
    a.x = xrow[ka];
    a.y = xrow[ka + 1];
    float w0 = W[(size_t)ka * kout + colc];
    float w1 = W[(size_t)(ka + 1) * kout + colc];
    b.x = colok ? w0 : 0.0f;
    b.y = colok ? w1 : 0.0f;
    acc = __builtin_amdgcn_wmma_f32_16x16x4_f32(false, a, false, b,
                                                (short)0, acc, false, false);
  }
  if constexpr ((KIN & 3) != 0) {        // compile-time tail (only KIN==3 here)
    int ka = KFULL + 2 * half;
    int ka0 = (ka     < KIN) ? ka     : (KIN - 1);
    int ka1 = (ka + 1 < KIN) ? ka + 1 : (KIN - 1);
    float xa0 = xrow[ka0];
    float xa1 = xrow[ka1];
    float w0  = W[(size_t)ka0 * kout + colc];
    float w1  = W[(size_t)ka1 * kout + colc];
    v2f a, b;
    a.x = (ka     < KIN) ? xa0 : 0.0f;
    a.y = (ka + 1 < KIN) ? xa1 : 0.0f;
    b.x = (ka     < KIN && colok) ? w0 : 0.0f;
    b.y = (ka + 1 < KIN && colok) ? w1 : 0.0f;
    acc = __builtin_amdgcn_wmma_f32_16x16x4_f32(false, a, false, b,
                                                (short)0, acc, false, false);
  }

  if (colok) {
    float bc = bias[col];
    int rbase = tileM * 16 + 8 * half;
#pragma unroll
    for (int r = 0; r < 8; ++r) {
      Y[(size_t)(rbase + r) * kout + col] = acc[r] + bc;
    }
  }
}

// alpha[e,h] = <q[dst,h,:], k[src,h,:] + e(edge,h,:)> / sqrt(C); atomic segment-max into amax[dst,h]
__global__ void edge_alpha_kernel(const float* __restrict__ Q,
                                  const float* __restrict__ Kb,
                                  const float* __restrict__ ea,
                                  const float* __restrict__ We,
                                  const float* __restrict__ be,
                                  const int* __restrict__ ei,
                                  float* __restrict__ alpha,
                                  float* __restrict__ amax,
                                  int C, int hc, float inv_sqrt_c) {
  int t = blockIdx.x * blockDim.x + threadIdx.x;
  if (t >= NEDGES * NHEADS) return;
  int e = t >> 2;
  int h = t & 3;
  int src = ei[e];
  int dst = ei[NEDGES + e];
  float a0 = ea[e * 3 + 0], a1 = ea[e * 3 + 1], a2 = ea[e * 3 + 2];
  const float* qrow = Q  + (size_t)dst * hc + h * C;
  const float* krow = Kb + (size_t)src * hc + h * C;
  float s = 0.0f;
  for (int c = 0; c < C; ++c) {
    int idx = h * C + c;
    float ev = a0 * We[idx] + a1 * We[hc + idx] + a2 * We[2 * hc + idx] + be[idx];
    s += qrow[c] * (krow[c] + ev);
  }
  s *= inv_sqrt_c;
  alpha[t] = s;
  atomicMaxF(&amax[dst * NHEADS + h], s);
}

// alpha[e,h] = exp(alpha - amax[dst,h]); atomic segment-sum into denom[dst,h]
__global__ void edge_exp_kernel(float* __restrict__ alpha,
                                const float* __restrict__ amax,
                                float* __restrict__ denom,
                                const int* __restrict__ ei) {
  int t = blockIdx.x * blockDim.x + threadIdx.x;
  if (t >= NEDGES * NHEADS) return;
  int e = t >> 2;
  int h = t & 3;
  int dst = ei[NEDGES + e];
  float ex = __expf(alpha[t] - amax[dst * NHEADS + h]);
  alpha[t] = ex;
  atomicAdd(&denom[dst * NHEADS + h], ex);
}

// O[dst,c] += (1/H) * sum_h softmax(alpha) * (v[src,h,c] + e(edge,h,c))
__global__ void edge_agg_kernel(const float* __restrict__ V,
                                const float* __restrict__ ea,
                                const float* __restrict__ We,
                                const float* __restrict__ be,
                                const float* __restrict__ alpha,
                                const float* __restrict__ denom,
                                const int* __restrict__ ei,
                                float* __restrict__ O,
                                int C, int hc) {
  int t = blockIdx.x * blockDim.x + threadIdx.x;
  if (t >= NEDGES * C) return;
  int e = t / C;
  int c = t - e * C;
  int src = ei[e];
  int dst = ei[NEDGES + e];
  float a0 = ea[e * 3 + 0], a1 = ea[e * 3 + 1], a2 = ea[e * 3 + 2];
  float acc = 0.0f;
#pragma unroll
  for (int h = 0; h < NHEADS; ++h) {
    int idx = h * C + c;
    float ev = a0 * We[idx] + a1 * We[hc + idx] + a2 * We[2 * hc + idx] + be[idx];
    float ve = V[(size_t)src * hc + idx] + ev;
    float an = alpha[e * NHEADS + h] / (denom[dst * NHEADS + h] + SM_EPS);
    acc += an * ve;
  }
  atomicAdd(&O[(size_t)dst * C + c], acc * 0.25f);
}

// beta-gated skip + sigmoid; optional BN statistics via atomics
__global__ void node_gate_kernel(const float* __restrict__ O,
                                 const float* __restrict__ R,
                                 const float* __restrict__ Wb,
                                 float* __restrict__ hout,
                                 int C,
                                 float* __restrict__ bnsum,
                                 float* __restrict__ bnssq,
                                 int doBN) {
  int n = blockIdx.x * blockDim.x + threadIdx.x;
  if (n >= NNODES) return;
  const float* o = O + (size_t)n * C;
  const float* r = R + (size_t)n * C;
  float d = 0.0f;
  for (int c = 0; c < C; ++c)
    d += Wb[c] * o[c] + Wb[C + c] * r[c] + Wb[2 * C + c] * (o[c] - r[c]);
  float g = sigmoidf_(d);
  for (int c = 0; c < C; ++c) {
    float hv = g * r[c] + (1.0f - g) * o[c];
    hv = sigmoidf_(hv);
    hout[(size_t)n * C + c] = hv;
    if (doBN) {
      atomicAdd(&bnsum[c], hv);
      atomicAdd(&bnssq[c], hv * hv);
    }
  }
}

__global__ void bn_apply_kernel(const float* __restrict__ hin,
                                float* __restrict__ hout,
                                const float* __restrict__ bnsum,
                                const float* __restrict__ bnssq,
                                const float* __restrict__ gamma,
                                const float* __restrict__ beta,
                                int C) {
  int t = blockIdx.x * blockDim.x + threadIdx.x;
  if (t >= NNODES * C) return;
  int c = t % C;
  float invN = 1.0f / (float)NNODES;
  float mu = bnsum[c] * invN;
  float var = bnssq[c] * invN - mu * mu;
  var = fmaxf(var, 0.0f);
  hout[t] = (hin[t] - mu) * rsqrtf(var + BN_EPS) * gamma[c] + beta[c];
}

__global__ void softmax2_kernel(const float* __restrict__ hin,
                                float* __restrict__ out) {
  int n = blockIdx.x * blockDim.x + threadIdx.x;
  if (n >= NNODES) return;
  float h0 = hin[n * 2 + 0];
  float h1 = hin[n * 2 + 1];
  float m = fmaxf(h0, h1);
  float e0 = __expf(h0 - m);
  float e1 = __expf(h1 - m);
  float inv = 1.0f / (e0 + e1);
  out[n * 2 + 0] = e0 * inv;
  out[n * 2 + 1] = e1 * inv;
}

// ---------------- host launcher ----------------

extern "C" void kernel_launch(void* const* d_in, const int* in_sizes, int n_in,
                              void* d_out, int out_size, void* d_ws, size_t ws_size,
                              hipStream_t stream) {
  // setup_inputs() order: x, edge_index, edge_attr, params{layers[4]{Wq,bq,Wk,bk,Wv,bv,We,be,Ws,bs,Wb}, bn[3]{gamma,beta}}
  const float* x  = (const float*)d_in[0];
  const int*   ei = (const int*)  d_in[1];   // [2, E] int32
  const float* ea = (const float*)d_in[2];   // [E, 3]
  auto P   = [&](int l, int j) { return (const float*)d_in[3 + 11 * l + j]; };
  auto BNp = [&](int b, int j) { return (const float*)d_in[3 + 44 + 2 * b + j]; }; // j: 0=gamma,1=beta

  // workspace layout (fp32), ~88 MB total
  float* ws    = (float*)d_ws;
  float* Q     = ws;                       // N*256
  float* Kb    = Q     + (size_t)NNODES * 256;
  float* V     = Kb    + (size_t)NNODES * 256;
  float* R     = V     + (size_t)NNODES * 256;   // N*64
  float* O     = R     + (size_t)NNODES * 64;    // N*64
  float* alpha = O     + (size_t)NNODES * 64;    // E*4
  float* amax  = alpha + (size_t)NEDGES * 4;     // N*4
  float* denom = amax  + (size_t)NNODES * 4;     // N*4
  float* bnsum = denom + (size_t)NNODES * 4;     // 64
  float* bnssq = bnsum + 64;                     // 64
  float* hA    = bnssq + 64;                     // N*64
  float* hB    = hA    + (size_t)NNODES * 64;    // N*64

  const int cins[4]  = {3, 64, 32, 16};
  const int couts[4] = {64, 32, 16, 2};
  const int tilesM = NNODES / 16;  // 1250

  auto launch_gemm = [&](const float* X, const float* W, const float* b, float* Y,
                         int cin, int kout) {
    int tilesN = (kout + 15) / 16;
    int nt = tilesM * tilesN;
    int blocks = (nt * 32 + 255) / 256;
    switch (cin) {
      case 3:  gemm_bias_wmma<3><<<blocks, 256, 0, stream>>>(X, W, b, Y, kout, tilesN, nt); break;
      case 16: gemm_bias_wmma<16><<<blocks, 256, 0, stream>>>(X, W, b, Y, kout, tilesN, nt); break;
      case 32: gemm_bias_wmma<32><<<blocks, 256, 0, stream>>>(X, W, b, Y, kout, tilesN, nt); break;
      default: gemm_bias_wmma<64><<<blocks, 256, 0, stream>>>(X, W, b, Y, kout, tilesN, nt); break;
    }
  };

  const float* xin = x;
  for (int l = 0; l < 4; ++l) {
    int cin = cins[l], C = couts[l], hc = 4 * C;

    // node GEMMs (fp32 WMMA)
    launch_gemm(xin, P(l,0), P(l,1), Q,  cin, hc);
    launch_gemm(xin, P(l,2), P(l,3), Kb, cin, hc);
    launch_gemm(xin, P(l,4), P(l,5), V,  cin, hc);
    launch_gemm(xin, P(l,8), P(l,9), R,  cin, C);

    // init reduction buffers
    fill_kernel<<<(NNODES * 4 + 255) / 256, 256, 0, stream>>>(amax, -INFINITY, NNODES * 4);
    fill_kernel<<<(NNODES * 4 + 255) / 256, 256, 0, stream>>>(denom, 0.0f, NNODES * 4);
    fill_kernel<<<(NNODES * C + 255) / 256, 256, 0, stream>>>(O, 0.0f, NNODES * C);
    fill_kernel<<<1, 128, 0, stream>>>(bnsum, 0.0f, 64);
    fill_kernel<<<1, 128, 0, stream>>>(bnssq, 0.0f, 64);

    // edge attention (edge embedding recomputed inline: 3 MACs/elem, avoids 327MB [E,H,C] buffers)
    float isc = 1.0f / sqrtf((float)C);
    edge_alpha_kernel<<<(NEDGES * 4 + 255) / 256, 256, 0, stream>>>(Q, Kb, ea, P(l,6), P(l,7), ei, alpha, amax, C, hc, isc);
    edge_exp_kernel<<<(NEDGES * 4 + 255) / 256, 256, 0, stream>>>(alpha, amax, denom, ei);
    edge_agg_kernel<<<(NEDGES * C + 255) / 256, 256, 0, stream>>>(V, ea, P(l,6), P(l,7), alpha, denom, ei, O, C, hc);

    // gate + sigmoid (+ BN stats), then BN or final softmax
    int doBN = (l < 3) ? 1 : 0;
    node_gate_kernel<<<(NNODES + 255) / 256, 256, 0, stream>>>(O, R, P(l,10), hA, C, bnsum, bnssq, doBN);
    if (doBN) {
      bn_apply_kernel<<<(NNODES * C + 255) / 256, 256, 0, stream>>>(hA, hB, bnsum, bnssq, BNp(l,0), BNp(l,1), C);
      xin = hB;
    } else {
      softmax2_kernel<<<(NNODES + 255) / 256, 256, 0, stream>>>(hA, (float*)d_out);
    }
  }
}